// LambdaReturn_88691074662797
// MI455X (gfx1250) — compile-verified
//
#include <hip/hip_runtime.h>

// Lambda-return (reverse affine scan), T=1000, B=32768, fp32.
// Memory-bound (AI ~0.3 FLOP/B): no WMMA — optimize the data-movement path.
// 3-phase chunked scan (chunk affine maps -> tiny cross-chunk scan -> final
// emit) so we have ~2.5K waves of MLP; all bulk traffic is global_load_b128 /
// global_store_b128 with non-temporal policy (touch-once, working set > L2).
// Shapes are compile-time constants so the unrolled loads use immediate
// offsets off a single walking base pointer (minimal SALU/VALU per VMEM).

typedef float v4f __attribute__((ext_vector_type(4)));

#define T_DIM  1000
#define NCHUNK 10
#define CLEN   100                 // T_DIM / NCHUNK
#define B_DIM  32768
#define B4C    (B_DIM / 4)         // 8192 float4 columns; row stride 128 KiB

__device__ constexpr float GL = 0.99f * 0.95f;          // gamma * lambda
__device__ constexpr float G1 = 0.99f * (1.0f - 0.95f); // gamma * (1 - lambda)

__device__ __forceinline__ v4f nt_load(const v4f* __restrict__ p) {
    return __builtin_nontemporal_load(p);
}
__device__ __forceinline__ void nt_store(v4f* __restrict__ p, v4f x) {
    __builtin_nontemporal_store(x, p);
}

// Pass 1: per (chunk, 4-column) thread, compose the chunk's affine map:
//   carry_out = A + Bc * carry_in   (scanning t from chunk top downward)
__global__ void lr_pass1_chunk_affine(const v4f* __restrict__ r,
                                      const v4f* __restrict__ v,
                                      const v4f* __restrict__ m,
                                      v4f* __restrict__ Aout,
                                      v4f* __restrict__ Bout) {
    const int j = blockIdx.x * blockDim.x + threadIdx.x;   // [0, B4C)
    const int c = blockIdx.y;                              // [0, NCHUNK)
    const int tTop = c * CLEN + (CLEN - 1);
    const size_t base = (size_t)tTop * B4C + j;

    const v4f* __restrict__ pr = r + base;
    const v4f* __restrict__ pv = v + base;
    const v4f* __restrict__ pm = m + base;

    v4f A  = {0.0f, 0.0f, 0.0f, 0.0f};
    v4f Bc = {1.0f, 1.0f, 1.0f, 1.0f};

#pragma unroll 5
    for (int k = 0; k < CLEN; ++k) {
        v4f rv = nt_load(pr);
        v4f vv = nt_load(pv);
        v4f mv = nt_load(pm);
        pr -= B4C; pv -= B4C; pm -= B4C;
        v4f coef  = GL * mv;
        v4f delta = rv + (G1 * mv) * vv;
        A  = delta + coef * A;
        Bc = coef * Bc;
    }
    Aout[c * B4C + j] = A;
    Bout[c * B4C + j] = Bc;
}

// Pass 2: tiny per-column scan across the NCHUNK affine maps.
// Seed carry = values[T-1]; emit carry entering each chunk.
__global__ void lr_pass2_chunk_scan(const v4f* __restrict__ v,
                                    const v4f* __restrict__ A,
                                    const v4f* __restrict__ Bc,
                                    v4f* __restrict__ cin) {
    const int j = blockIdx.x * blockDim.x + threadIdx.x;

    v4f carry = nt_load(v + (size_t)(T_DIM - 1) * B4C + j);
#pragma unroll
    for (int c = NCHUNK - 1; c >= 0; --c) {
        cin[c * B4C + j] = carry;                            // carry entering chunk c
        carry = A[c * B4C + j] + Bc[c * B4C + j] * carry;    // carry leaving to chunk c-1
    }
}

// Pass 3: redo the in-chunk scan with the correct incoming carry and
// stream the results out with NT stores.
__global__ void lr_pass3_final(const v4f* __restrict__ r,
                               const v4f* __restrict__ v,
                               const v4f* __restrict__ m,
                               const v4f* __restrict__ cin,
                               v4f* __restrict__ out) {
    const int j = blockIdx.x * blockDim.x + threadIdx.x;
    const int c = blockIdx.y;
    const int tTop = c * CLEN + (CLEN - 1);
    const size_t base = (size_t)tTop * B4C + j;

    const v4f* __restrict__ pr = r + base;
    const v4f* __restrict__ pv = v + base;
    const v4f* __restrict__ pm = m + base;
    v4f*       __restrict__ po = out + base;

    v4f carry = cin[c * B4C + j];

#pragma unroll 5
    for (int k = 0; k < CLEN; ++k) {
        v4f rv = nt_load(pr);
        v4f vv = nt_load(pv);
        v4f mv = nt_load(pm);
        pr -= B4C; pv -= B4C; pm -= B4C;
        v4f coef  = GL * mv;
        v4f delta = rv + (G1 * mv) * vv;
        carry = delta + coef * carry;
        nt_store(po, carry);
        po -= B4C;
    }
}

extern "C" void kernel_launch(void* const* d_in, const int* in_sizes, int n_in,
                              void* d_out, int out_size, void* d_ws, size_t ws_size,
                              hipStream_t stream) {
    const v4f* r = (const v4f*)d_in[0];   // rewards (T,B) f32
    const v4f* v = (const v4f*)d_in[1];   // values  (T,B) f32
    const v4f* m = (const v4f*)d_in[2];   // masks   (T,B) f32

    // Workspace layout: A | Bc | carry-in, each NCHUNK*B floats (~3.9 MB total)
    float* ws  = (float*)d_ws;
    v4f*   A   = (v4f*)(ws);
    v4f*   Bc  = (v4f*)(ws + (size_t)NCHUNK * B_DIM);
    v4f*   cin = (v4f*)(ws + (size_t)2 * NCHUNK * B_DIM);

    dim3 blk(256);
    dim3 grid1(B4C / 256, NCHUNK);       // 32 x 10 workgroups
    dim3 grid2(B4C / 256);

    lr_pass1_chunk_affine<<<grid1, blk, 0, stream>>>(r, v, m, A, Bc);
    lr_pass2_chunk_scan<<<grid2, blk, 0, stream>>>(v, A, Bc, cin);
    lr_pass3_final<<<grid1, blk, 0, stream>>>(r, v, m, cin, (v4f*)d_out);
}